// TSF_72576357368148
// MI455X (gfx1250) — compile-verified
//
#include <hip/hip_runtime.h>

// MI455X / gfx1250, wave32. Memory-bound batched skinny GEMM:
// out[b,c,n] = sum_t f[b,n,t] * video[b,c,t], via v_wmma_f32_16x16x4_f32
// (pure f32 datapath: no precision loss, no conversion VALU work).

typedef __attribute__((ext_vector_type(8))) float v8f;
typedef __attribute__((ext_vector_type(2))) float v2f;

#define B_  32
#define C_  1024
#define T_  1024
#define N_  3
#define FS  1032   // 1024 + 8 floats: keeps 8B alignment, skews LDS banks between rows

__global__ __launch_bounds__(128) void tsf_wmma_kernel(
    const float* __restrict__ video,   // (B, C, T)
    const int*   __restrict__ length,  // (B,)
    const float* __restrict__ center,  // (N,)
    const float* __restrict__ gamma_,  // (N,)
    float*       __restrict__ out)     // (B, C*N)
{
  // rows 0..2: normalized filter bank f[n][t] (f32); row 3: zeros (B cols 3..15)
  __shared__ __align__(16) float s_f[(N_ + 1) * FS];   // ~16.5 KB
  __shared__ float s_sum[N_];

  const int b   = blockIdx.y;
  const int tid = threadIdx.x;

  // ---------- phase 1: per-batch normalized Cauchy filter bank ----------
  const float lf = (float)length[b];
  float ctr[N_], inv_g[N_], amp[N_];
#pragma unroll
  for (int n = 0; n < N_; ++n) {
    float cn = tanhf(center[n]);
    float gn = tanhf(gamma_[n]);
    float g  = expf(1.5f - 2.0f * fabsf(gn));
    ctr[n]   = (lf - 1.0f) * (cn + 1.0f) * 0.5f;
    inv_g[n] = 1.0f / g;
    amp[n]   = 1.0f / (3.14159265358979323846f * g);
  }
  if (tid < N_) s_sum[tid] = 0.0f;
  __syncthreads();

  float ls[N_] = {0.0f, 0.0f, 0.0f};
  for (int t = tid; t < T_; t += 128) {
    float tf = (float)t;
#pragma unroll
    for (int n = 0; n < N_; ++n) {
      float d = (tf - ctr[n]) * inv_g[n];
      float w = amp[n] / (1.0f + d * d);
      s_f[n * FS + t] = w;
      ls[n] += w;
    }
  }
#pragma unroll
  for (int n = 0; n < N_; ++n) atomicAdd(&s_sum[n], ls[n]);   // ds_add_f32
  __syncthreads();

  float inv_sum[N_];
#pragma unroll
  for (int n = 0; n < N_; ++n) inv_sum[n] = 1.0f / (s_sum[n] + 1e-6f);
  for (int t = tid; t < T_; t += 128) {
#pragma unroll
    for (int n = 0; n < N_; ++n) s_f[n * FS + t] *= inv_sum[n];
  }
  for (int t = tid; t < FS; t += 128) s_f[N_ * FS + t] = 0.0f;  // zero row
  __syncthreads();

  // ---------- phase 2: per-wave 16-channel tile, f32 WMMA over T ----------
  const int wave = tid >> 5;
  const int lane = tid & 31;
  const int hi   = lane >> 4;   // K-half selector (ISA 16x4 A / 4x16 B layout)
  const int lo   = lane & 15;   // A: row M ; B: col N ; C/D: col N
  const int c0   = blockIdx.x * 64 + wave * 16;

  // A 16x4 f32: lanes 0-15 hold {K0,K1}, lanes 16-31 hold {K2,K3}, row M = lo.
  const float* rp = video + ((size_t)b * C_ + (size_t)(c0 + lo)) * (size_t)T_ + 2 * hi;
  // B 4x16 f32: lane col = lo (zeros row for lo>=3), same K-half split.
  const float* fp = &s_f[((lo < N_) ? lo : N_) * FS + 2 * hi];

  v8f acc = {};
#pragma unroll 8
  for (int t = 0; t < T_; t += 4) {
    v2f a  = __builtin_nontemporal_load((const v2f*)(rp + t));
    v2f bb = *(const v2f*)(fp + t);
    acc = __builtin_amdgcn_wmma_f32_16x16x4_f32(
        false, a, false, bb, (short)0, acc, false, false);
  }

  // C/D layout: lane L -> N = L&15, M = vgpr + (L>>4)*8. Only N = 0..2 are real.
  if (lo < N_) {
    float* outb = out + (size_t)b * (size_t)(C_ * N_);
#pragma unroll
    for (int r = 0; r < 8; ++r) {
      int m = r + hi * 8;
      outb[(size_t)(c0 + m) * N_ + lo] = acc[r];
    }
  }
}

extern "C" void kernel_launch(void* const* d_in, const int* in_sizes, int n_in,
                              void* d_out, int out_size, void* d_ws, size_t ws_size,
                              hipStream_t stream) {
  (void)in_sizes; (void)n_in; (void)d_ws; (void)ws_size; (void)out_size;
  const float* video  = (const float*)d_in[0];
  const int*   length = (const int*)d_in[1];
  const float* center = (const float*)d_in[2];
  // d_in[3] = delta : multiplied by zero in the reference, no effect on output
  const float* gamma_ = (const float*)d_in[4];
  float* out = (float*)d_out;

  dim3 grid(C_ / 64, B_);   // 16 x 32 = 512 blocks, 4 waves each
  dim3 block(128);
  hipLaunchKernelGGL(tsf_wmma_kernel, grid, block, 0, stream,
                     video, length, center, gamma_, out);
}